// SimpleGeoBloom_82214263980289
// MI455X (gfx1250) — compile-verified
//
#include <hip/hip_runtime.h>
#include <hip/hip_bf16.h>
#include <math.h>

// Problem constants (from reference)
#define Qn   48
#define Pn   384
#define Dn   8192
#define H1n  128
#define H2n  32
#define NROWS 432          // Qn + Pn  (== 27 * 16, WMMA-tile exact)
#define NW   256           // Dn / 32 bitmask words per row
#define SLOPE 0.125f
#define KWAVES 8           // K-split waves per embed-GEMM block

typedef __attribute__((ext_vector_type(2))) float v2f;
typedef __attribute__((ext_vector_type(8))) float v8f;

// ---------------------------------------------------------------------------
// WMMA helper: D = A(16x4) x B(4x16) + C, fp32 exact.
// A layout (ISA 7.12.2, 32-bit A 16x4): lanes 0-15 hold M=lane with
//   VGPR0=K0,VGPR1=K1; lanes 16-31 hold same M with K2,K3.
// B layout mirrors A with N across lanes (B[k][n], n = lane%16,
//   k = 2*(lane>=16) + vgpr).
// C/D (32-bit 16x16): VGPR r -> M = r + 8*(lane>=16), N = lane%16.
// Per k-step ks, every lane loads k = 4*ks + 2*(lane>>4) + {0,1} for both
// A (its M-row) and B (its N-column of the left operand's transpose).
// ---------------------------------------------------------------------------
__device__ __forceinline__ v8f wmma_f32_4(v2f a, v2f b, v8f c) {
  return __builtin_amdgcn_wmma_f32_16x16x4_f32(false, a, false, b, (short)0, c,
                                               false, false);
}

// ---------------------------------------------------------------------------
// Kernel 1: per-row popcount (query_bits / poi_bits) + pack 0/1 floats into
// bitmasks (256 u32 words per row). One block per row, 256 threads.
// ---------------------------------------------------------------------------
__global__ void k_bits_mask(const float* __restrict__ qd,
                            const float* __restrict__ px,
                            float* __restrict__ bits,
                            unsigned* __restrict__ masks) {
  const int row = blockIdx.x;
  const int t = threadIdx.x;  // 0..255, each owns 32 consecutive elements
  const float* src =
      (row < Qn) ? (qd + (size_t)row * Dn) : (px + (size_t)(row - Qn) * Dn);
  const float* base = src + t * 32;
  unsigned w = 0;
  int cnt = 0;
#pragma unroll
  for (int b = 0; b < 32; ++b) {
    if (base[b] != 0.0f) { w |= (1u << b); ++cnt; }
  }
  masks[(size_t)row * NW + t] = w;

  __shared__ int s[256];
  s[t] = cnt;
  __syncthreads();
  for (int off = 128; off > 0; off >>= 1) {
    if (t < off) s[t] += s[t + off];
    __syncthreads();
  }
  if (t == 0) bits[row] = (float)s[0];
}

// ---------------------------------------------------------------------------
// Kernel 2: embedding GEMM with in-block K-split.
//   R[m][n] = relu((X[m,:] . enc_W[n,:] + b[n]) / bits[m])
// X = concat(query_dense, poi_x) [432 x 8192], enc_W [128 x 8192].
// Grid (27 mtiles, 8 ntiles); block = 256 threads = 8 waves, each wave
// accumulates a K=1024 slice (256 WMMAs), partials reduced via LDS by wave 0.
// 1728 waves total (vs 216 before) to fill the machine; deterministic
// (no float atomics) so graph replays are bit-stable.
// global_prefetch_b8 issued two 128B lines ahead of the load stream.
// ---------------------------------------------------------------------------
__global__ void k_embed_gemm(const float* __restrict__ qd,
                             const float* __restrict__ px,
                             const float* __restrict__ encW,
                             const float* __restrict__ encb,
                             const float* __restrict__ bits,
                             float* __restrict__ R) {
  const int lane = threadIdx.x & 31;
  const int wave = threadIdx.x >> 5;   // 0..7 (K-slice owner)
  const int mtile = blockIdx.x;        // 0..26
  const int ntile = blockIdx.y;        // 0..7
  const int mloc = lane & 15;
  const int khalf = lane >> 4;         // 0 or 1

  const int m = mtile * 16 + mloc;
  const int n = ntile * 16 + mloc;
  const float* arow =
      (m < Qn) ? (qd + (size_t)m * Dn) : (px + (size_t)(m - Qn) * Dn);
  const float* brow = encW + (size_t)n * Dn;

  const int kbase = wave * (Dn / KWAVES);
  const int kend = kbase + (Dn / KWAVES);

  v8f acc = {};
  // 32 floats (one 128B cacheline) per outer iteration; prefetch 2 lines out.
  for (int k0 = kbase; k0 < kend; k0 += 32) {
    __builtin_prefetch(arow + k0 + 64, 0, 3);
    __builtin_prefetch(brow + k0 + 64, 0, 3);
#pragma unroll
    for (int k = k0; k < k0 + 32; k += 4) {
      const int kk = k + 2 * khalf;
      v2f a; a.x = arow[kk]; a.y = arow[kk + 1];
      v2f b; b.x = brow[kk]; b.y = brow[kk + 1];
      acc = wmma_f32_4(a, b, acc);
    }
  }

  // LDS reduction of the 8 per-wave partial tiles (linear index r*32+lane,
  // identical lane mapping on writer and reader, so no m/n decode needed).
  __shared__ float red[KWAVES - 1][8 * 32];
  if (wave > 0) {
#pragma unroll
    for (int r = 0; r < 8; ++r) red[wave - 1][r * 32 + lane] = acc[r];
  }
  __syncthreads();
  if (wave == 0) {
#pragma unroll
    for (int r = 0; r < 8; ++r) {
      float v = acc[r];
      for (int w = 0; w < KWAVES - 1; ++w) v += red[w][r * 32 + lane];
      acc[r] = v;
    }
    const int nout = ntile * 16 + (lane & 15);
    const float bv = encb[nout];
#pragma unroll
    for (int r = 0; r < 8; ++r) {
      const int mout = mtile * 16 + r + 8 * (lane >> 4);
      float v = (acc[r] + bv) / bits[mout];
      v = v > 0.0f ? v : 0.0f;  // relu(all_embed) fused here
      R[(size_t)mout * H1n + nout] = v;
    }
  }
}

// ---------------------------------------------------------------------------
// Kernel 3: W1 split GEMM.  AB[m][n] = R[m,:] . W1[n, off:off+128]
// where off=0 for query rows (mtiles 0..2) and off=128 for poi rows.
// Grid (27 mtiles, 2 ntiles), one wave per block, K=128.
// ---------------------------------------------------------------------------
__global__ void k_w1_gemm(const float* __restrict__ R,
                          const float* __restrict__ W1,
                          float* __restrict__ AB) {
  const int lane = threadIdx.x;
  const int mtile = blockIdx.x;  // 0..26
  const int ntile = blockIdx.y;  // 0..1
  const int mloc = lane & 15;
  const int khalf = lane >> 4;

  const int m = mtile * 16 + mloc;
  const int n = ntile * 16 + mloc;
  const int colOff = (mtile * 16 < Qn) ? 0 : H1n;  // W1a vs W1b
  const float* arow = R + (size_t)m * H1n;
  const float* brow = W1 + (size_t)n * (2 * H1n) + colOff;

  v8f acc = {};
#pragma unroll
  for (int k = 0; k < H1n; k += 4) {
    const int kk = k + 2 * khalf;
    v2f a; a.x = arow[kk]; a.y = arow[kk + 1];
    v2f b; b.x = brow[kk]; b.y = brow[kk + 1];
    acc = wmma_f32_4(a, b, acc);
  }

  const int nout = ntile * 16 + (lane & 15);
#pragma unroll
  for (int r = 0; r < 8; ++r) {
    const int mout = mtile * 16 + r + 8 * (lane >> 4);
    AB[(size_t)mout * H2n + nout] = acc[r];
  }
}

// ---------------------------------------------------------------------------
// Kernel 4: per-pair tile. Each wave owns 16 pairs (same q, 16 consecutive p).
//  h1[i][k] = relu(A[q][k] + B[p_i][k] + b1[k])   built on the fly as WMMA A
//  h2 = relu(h1 @ W2^T + b2)  via 2 n-tiles x 8 k-steps = 16 WMMAs
//  desc_raw[q][p] = sum over d in (qmask & pmask) of leaky(W3[d].h2)+1, /qbits
// Grid 1152 (= 48*384/16) blocks, 32 threads.
// ---------------------------------------------------------------------------
__global__ void k_pair(const float* __restrict__ AB,
                       const float* __restrict__ b1,
                       const float* __restrict__ W2,
                       const float* __restrict__ b2,
                       const float* __restrict__ W3,
                       const unsigned* __restrict__ masks,
                       const float* __restrict__ bits,
                       float* __restrict__ desc_raw) {
  const int lane = threadIdx.x;
  const int tile = blockIdx.x;           // 0..1151
  const int q = tile / (Pn / 16);        // Pn/16 == 24 tiles per q row
  const int pbase = (tile % (Pn / 16)) * 16;
  const int iloc = lane & 15;
  const int khalf = lane >> 4;

  const float* Aq = AB + (size_t)q * H2n;              // query row of AB
  const float* Bp = AB + (size_t)(Qn + pbase) * H2n;   // poi rows of AB

  v8f acc0 = {}, acc1 = {};
#pragma unroll
  for (int ks = 0; ks < 8; ++ks) {
    const int k = ks * 4 + 2 * khalf;
    // A-matrix element: h1 row iloc, cols k,k+1 (relu(A[q]+B[p]+b1))
    const float* br = Bp + (size_t)iloc * H2n;
    float x0 = Aq[k] + br[k] + b1[k];
    float x1 = Aq[k + 1] + br[k + 1] + b1[k + 1];
    v2f a; a.x = x0 > 0.0f ? x0 : 0.0f; a.y = x1 > 0.0f ? x1 : 0.0f;
    // B-matrix: W2^T, two 16-wide n tiles.  B[k][n] = W2[n][k]
    v2f bm0, bm1;
    bm0.x = W2[(size_t)iloc * H2n + k];
    bm0.y = W2[(size_t)iloc * H2n + k + 1];
    bm1.x = W2[(size_t)(16 + iloc) * H2n + k];
    bm1.y = W2[(size_t)(16 + iloc) * H2n + k + 1];
    acc0 = wmma_f32_4(a, bm0, acc0);
    acc1 = wmma_f32_4(a, bm1, acc1);
  }

  // h2 tile -> LDS with bias + relu
  __shared__ float h2s[16][H2n];
  {
    const int n0 = lane & 15;
    const int ioff = 8 * (lane >> 4);
#pragma unroll
    for (int r = 0; r < 8; ++r) {
      const int i = r + ioff;
      float v0 = acc0[r] + b2[n0];
      float v1 = acc1[r] + b2[16 + n0];
      h2s[i][n0] = v0 > 0.0f ? v0 : 0.0f;
      h2s[i][16 + n0] = v1 > 0.0f ? v1 : 0.0f;
    }
  }
  __syncthreads();

  // Sparse masked W3 reduction: only d where both bitmaps have a 1.
  const unsigned* qm = masks + (size_t)q * NW;
  const float qb = bits[q];
  for (int i = 0; i < 16; ++i) {
    const int p = pbase + i;
    const unsigned* pm = masks + (size_t)(Qn + p) * NW;
    float part = 0.0f;
    for (int w = lane; w < NW; w += 32) {
      unsigned inter = qm[w] & pm[w];
      while (inter) {
        const int b = __ffs(inter) - 1;
        inter &= inter - 1u;
        const int d = w * 32 + b;
        const float* w3 = W3 + (size_t)d * H2n;
        float dot = 0.0f;
#pragma unroll
        for (int kk = 0; kk < H2n; ++kk) dot = fmaf(w3[kk], h2s[i][kk], dot);
        part += (dot >= 0.0f ? dot : dot * SLOPE) + 1.0f;  // leaky(1/8)+1
      }
    }
    // wave32 butterfly reduce
#pragma unroll
    for (int off = 16; off > 0; off >>= 1) part += __shfl_xor(part, off, 32);
    if (lane == 0) desc_raw[(size_t)q * Pn + p] = part / qb;
  }
}

// ---------------------------------------------------------------------------
// Kernel 5: per-q-row normalization (mean, unbiased std) + geo score.
// 48 blocks x 384 threads (one thread per p).
// ---------------------------------------------------------------------------
__global__ void k_finalize(const float* __restrict__ desc_raw,
                           const float* __restrict__ qloc,
                           const float* __restrict__ ploc,
                           const float* __restrict__ pa,
                           const float* __restrict__ pb,
                           const float* __restrict__ pc,
                           const float* __restrict__ pd,
                           float* __restrict__ out) {
  const int q = blockIdx.x;
  const int p = threadIdx.x;  // 0..383
  const float v = desc_raw[(size_t)q * Pn + p];

  __shared__ float ssum[512];
  __shared__ float ssq[512];
  ssum[p] = v;
  ssq[p] = v * v;
  if (p < 512 - Pn) { ssum[Pn + p] = 0.0f; ssq[Pn + p] = 0.0f; }
  __syncthreads();
  for (int off = 256; off > 0; off >>= 1) {
    if (p < off) { ssum[p] += ssum[p + off]; ssq[p] += ssq[p + off]; }
    __syncthreads();
  }
  const float mean = ssum[0] / (float)Pn;
  float var = (ssq[0] - (float)Pn * mean * mean) / (float)(Pn - 1);
  var = var > 0.0f ? var : 0.0f;
  const float z = (v - mean) / (sqrtf(var) + 1e-6f);

  const float dx = qloc[q * 2 + 0] - ploc[p * 2 + 0];
  const float dy = qloc[q * 2 + 1] - ploc[p * 2 + 1];
  const float dist_sim = -logf(sqrtf(dx * dx + dy * dy) + 1.0f);
  const float sig = 1.0f / (1.0f + expf(-(pa[0] * z + pb[0])));
  out[(size_t)q * Pn + p] = (pc[0] - sig) * (dist_sim - pd[0]);
}

// ---------------------------------------------------------------------------
// Launcher
// ---------------------------------------------------------------------------
extern "C" void kernel_launch(void* const* d_in, const int* in_sizes, int n_in,
                              void* d_out, int out_size, void* d_ws,
                              size_t ws_size, hipStream_t stream) {
  const float* qd   = (const float*)d_in[0];   // [48,8192]
  const float* qloc = (const float*)d_in[1];   // [48,2]
  const float* px   = (const float*)d_in[2];   // [384,8192]
  const float* ploc = (const float*)d_in[3];   // [384,2]
  const float* encW = (const float*)d_in[4];   // [128,8192]
  const float* encb = (const float*)d_in[5];   // [128]
  const float* W1   = (const float*)d_in[6];   // [32,256]
  const float* b1   = (const float*)d_in[7];   // [32]
  const float* W2   = (const float*)d_in[8];   // [32,32]
  const float* b2   = (const float*)d_in[9];   // [32]
  const float* W3   = (const float*)d_in[10];  // [8192,32]
  const float* pa   = (const float*)d_in[11];
  const float* pb   = (const float*)d_in[12];
  const float* pc   = (const float*)d_in[13];
  const float* pd   = (const float*)d_in[14];
  float* out = (float*)d_out;                  // [48,384]

  // Workspace carve-up (floats / u32, 4B aligned): ~0.8 MB total
  float* ws = (float*)d_ws;
  float* bits      = ws;                                  // 432
  unsigned* masks  = (unsigned*)(ws + NROWS);             // 432*256 u32
  float* R         = ws + NROWS + NROWS * NW;             // 432*128
  float* AB        = R + (size_t)NROWS * H1n;             // 432*32
  float* desc_raw  = AB + (size_t)NROWS * H2n;            // 48*384

  k_bits_mask<<<NROWS, 256, 0, stream>>>(qd, px, bits, masks);
  k_embed_gemm<<<dim3(NROWS / 16, H1n / 16), 32 * KWAVES, 0, stream>>>(
      qd, px, encW, encb, bits, R);
  k_w1_gemm<<<dim3(NROWS / 16, H2n / 16), 32, 0, stream>>>(R, W1, AB);
  k_pair<<<(Qn * Pn) / 16, 32, 0, stream>>>(AB, b1, W2, b2, W3, masks, bits,
                                            desc_raw);
  k_finalize<<<Qn, Pn, 0, stream>>>(desc_raw, qloc, ploc, pa, pb, pc, pd, out);
}